// LatentLayer_88441966559687
// MI455X (gfx1250) — compile-verified
//
#include <hip/hip_runtime.h>
#include <math.h>

// Problem constants (match reference)
#define NV 32768
#define MV 8192
#define ZV 256

// Tiling
#define TM   128                 // columns (m) per workgroup
#define BN   128                 // rows (n) per n-block
#define KS   16                  // k-slice width staged in LDS (double-buffered)
#define LDA  20                  // padded LDS row stride (floats): 16B-aligned rows, conflict-free
#define NCH  8                   // N split into 8 chunks of 4096
#define CHROWS (NV / NCH)
#define NSLICE (ZV / KS)

typedef float v2f __attribute__((ext_vector_type(2)));
typedef float v8f __attribute__((ext_vector_type(8)));
typedef int   v4i __attribute__((ext_vector_type(4)));

typedef __attribute__((address_space(1))) v4i* gvec4p;   // global <4 x i32>*
typedef __attribute__((address_space(3))) v4i* lvec4p;   // LDS    <4 x i32>*

#if __has_builtin(__builtin_amdgcn_global_load_async_to_lds_b128) && \
    __has_builtin(__builtin_amdgcn_s_wait_asynccnt)
#define USE_ASYNC_LDS 1
#else
#define USE_ASYNC_LDS 0
#endif

// Stage a 128-row x KS-float slice (row-major, padded stride LDA) into LDS.
// 128 * 16 floats = 512 float4s -> 2 per thread (256 threads).
__device__ __forceinline__ void stage_rows(const float* __restrict__ src,
                                           float* __restrict__ dst,
                                           int tid, int kk)
{
    #pragma unroll
    for (int q = 0; q < 2; ++q) {
        int idx = tid + q * 256;
        int r   = idx >> 2;          // 0..127
        int c4  = (idx & 3) * 4;     // 0,4,8,12
#if USE_ASYNC_LDS
        float* srcnc = const_cast<float*>(src);
        __builtin_amdgcn_global_load_async_to_lds_b128(
            (gvec4p)&srcnc[(size_t)r * ZV + kk + c4],
            (lvec4p)&dst[r * LDA + c4],
            /*offset=*/0, /*cpol=*/0);
#else
        float4 v = *(const float4*)&src[(size_t)r * ZV + kk + c4];
        *(float4*)&dst[r * LDA + c4] = v;
#endif
    }
}

__global__ __launch_bounds__(256)
void latent_lse_gemm_kernel(const float* __restrict__ mean,
                            const float* __restrict__ e,
                            const float* __restrict__ sigma,
                            float* __restrict__ wsMax,
                            float* __restrict__ wsSum)
{
    __shared__ float sA[2][BN * LDA];   // mean slice  [buf][row n][k]
    __shared__ float sB[2][TM * LDA];   // e slice     [buf][col m][k]
    __shared__ float sA2[BN];           // ||mean_n||^2 partials for this n-block

    const int tid  = threadIdx.x;
    const int wave = tid >> 5;
    const int lane = tid & 31;
    const int lrow = lane & 15;          // row/col within 16
    const int lkk  = (lane >> 4) * 2;    // k sub-offset per lane half
    const int half = lane >> 4;          // 0 or 1 (C/D row offset of 8)

    const int mBase = blockIdx.x * TM;
    const int nBase = blockIdx.y * CHROWS;

    const float sg = sigma[0];
    const float c  = -1.0f / (2.0f * sg * sg);
    const float LOG2E = 1.4426950408889634f;

    const float* gB = e + (size_t)mBase * ZV;   // constant per workgroup

    // Online logsumexp state (base-2 domain) for column mBase + wave*16 + lrow.
    float runM = -INFINITY;
    float runS = 0.0f;

    for (int nb = 0; nb < CHROWS; nb += BN) {
        const float* gA = mean + (size_t)(nBase + nb) * ZV;

        // accumulators: 8 n-subtiles x (16x16 f32 tile = v8f)
        v8f D[8];
        #pragma unroll
        for (int t = 0; t < 8; ++t)
            #pragma unroll
            for (int i = 0; i < 8; ++i) D[t][i] = 0.0f;

        __syncthreads();                 // previous block's readers of sA2/buffers done
        if (tid < BN) sA2[tid] = 0.0f;

        // Prologue: issue fill of buffer 0 (slice 0)
        stage_rows(gA, sA[0], tid, 0);
        stage_rows(gB, sB[0], tid, 0);

        for (int s = 0; s < NSLICE; ++s) {
            const int   cur = s & 1;
            const int   kk  = s * KS;
            const float* bufA = sA[cur];
            const float* bufB = sB[cur];

#if USE_ASYNC_LDS
            __builtin_amdgcn_s_wait_asynccnt(0);   // this wave's async fills landed
#endif
            __syncthreads();                       // all waves' fills landed; prev readers done

            // Issue next slice's fill into the other buffer while we compute.
            if (s + 1 < NSLICE) {
                stage_rows(gA, sA[cur ^ 1], tid, kk + KS);
                stage_rows(gB, sB[cur ^ 1], tid, kk + KS);
            }

            // Row-norm partials as a by-product of staged A
            if (tid < BN) {
                float acc = 0.0f;
                #pragma unroll
                for (int k = 0; k < KS; ++k) {
                    float x = bufA[tid * LDA + k];
                    acc += x * x;
                }
                sA2[tid] += acc;
            }

            // WMMA over this k-slice: KS/4 steps of K=4
            #pragma unroll
            for (int ks = 0; ks < KS; ks += 4) {
                v2f bf = *(const v2f*)&bufB[(wave * 16 + lrow) * LDA + ks + lkk];
                #pragma unroll
                for (int t = 0; t < 8; ++t) {
                    v2f af = *(const v2f*)&bufA[(t * 16 + lrow) * LDA + ks + lkk];
                    D[t] = __builtin_amdgcn_wmma_f32_16x16x4_f32(
                        false, af, false, bf, (short)0, D[t], false, false);
                }
            }
        }
        __syncthreads();                 // sA2 complete before epilogue reads

        // Online LSE update per 16x16 tile (reduce over the n/M dimension)
        #pragma unroll
        for (int t = 0; t < 8; ++t) {
            float h[8];
            float tmx = -INFINITY;
            #pragma unroll
            for (int v = 0; v < 8; ++v) {
                int r = t * 16 + v + half * 8;
                float g = c * (sA2[r] - 2.0f * D[t][v]);   // c*a2[n] - 2c*ab
                h[v] = g * LOG2E;                          // base-2 domain
                tmx = fmaxf(tmx, h[v]);
            }
            tmx = fmaxf(tmx, __shfl_xor(tmx, 16, 32));     // merge lane halves
            float nm = fmaxf(runM, tmx);
            runS *= exp2f(runM - nm);
            float p = 0.0f;
            #pragma unroll
            for (int v = 0; v < 8; ++v) p += exp2f(h[v] - nm);
            p += __shfl_xor(p, 16, 32);
            runS += p;
            runM = nm;
        }
    }

    // Write per-(chunk, column) partials; lanes 0..15 hold the 16 columns.
    if (lane < 16) {
        int m = mBase + wave * 16 + lane;
        wsMax[(size_t)blockIdx.y * MV + m] = runM;
        wsSum[(size_t)blockIdx.y * MV + m] = runS;
    }
}

__global__ __launch_bounds__(256)
void latent_lse_combine_kernel(const float* __restrict__ e,
                               const float* __restrict__ sigma,
                               const float* __restrict__ wsMax,
                               const float* __restrict__ wsSum,
                               float* __restrict__ partial)
{
    const int m = blockIdx.x * 256 + threadIdx.x;
    const float sg = sigma[0];
    const float c  = -1.0f / (2.0f * sg * sg);

    float H = -INFINITY;
    #pragma unroll
    for (int k = 0; k < NCH; ++k)
        H = fmaxf(H, wsMax[(size_t)k * MV + m]);
    float S = 0.0f;
    #pragma unroll
    for (int k = 0; k < NCH; ++k)
        S += wsSum[(size_t)k * MV + m] * exp2f(wsMax[(size_t)k * MV + m] - H);

    float lse = (H + log2f(S)) * 0.69314718055994531f;     // back to natural log

    // b2[m] = ||e_m||^2
    float b2 = 0.0f;
    for (int k = 0; k < ZV; k += 4) {
        float4 v = *(const float4*)&e[(size_t)m * ZV + k];
        b2 += v.x * v.x + v.y * v.y + v.z * v.z + v.w * v.w;
    }
    float val = lse + c * b2;

    __shared__ float red[256];
    red[threadIdx.x] = val;
    __syncthreads();
    for (int s = 128; s > 0; s >>= 1) {
        if (threadIdx.x < s) red[threadIdx.x] += red[threadIdx.x + s];
        __syncthreads();
    }
    if (threadIdx.x == 0) partial[blockIdx.x] = red[0];
}

__global__ void latent_lse_final_kernel(const float* __restrict__ partial,
                                        float* __restrict__ out)
{
    float v = partial[threadIdx.x];        // 32 partials, one wave
    #pragma unroll
    for (int o = 16; o > 0; o >>= 1) v += __shfl_xor(v, o, 32);
    if (threadIdx.x == 0) out[0] = -v / (float)MV;
}

extern "C" void kernel_launch(void* const* d_in, const int* in_sizes, int n_in,
                              void* d_out, int out_size, void* d_ws, size_t ws_size,
                              hipStream_t stream)
{
    const float* mean  = (const float*)d_in[0];
    const float* e     = (const float*)d_in[1];
    const float* sigma = (const float*)d_in[2];
    float* out = (float*)d_out;

    float* ws      = (float*)d_ws;
    float* wsMax   = ws;                       // NCH * MV floats
    float* wsSum   = wsMax + (size_t)NCH * MV; // NCH * MV floats
    float* partial = wsSum + (size_t)NCH * MV; // MV/256 floats

    dim3 gridA(MV / TM, NCH);
    latent_lse_gemm_kernel<<<gridA, 256, 0, stream>>>(mean, e, sigma, wsMax, wsSum);
    latent_lse_combine_kernel<<<MV / 256, 256, 0, stream>>>(e, sigma, wsMax, wsSum, partial);
    latent_lse_final_kernel<<<1, 32, 0, stream>>>(partial, out);
}